// GCN_40080634806825
// MI455X (gfx1250) — compile-verified
//
#include <hip/hip_runtime.h>
#include <hip/hip_bf16.h>

typedef __attribute__((ext_vector_type(2))) float v2f;
typedef __attribute__((ext_vector_type(8))) float v8f;

// ---------------------------------------------------------------------------
// GEMM: Y[N x COUT] = X[N x 128] * W[128 x COUT] (+ bias), fp32 WMMA 16x16x4.
// Block covers 32 rows x COUT cols; each wave owns one 16-col stripe and two
// 16x16 row tiles (two independent accumulator chains reusing one B frag).
// A panel staged in LDS row-stride 132; W staged TRANSPOSED row-stride 134 so
// B frags are single aligned ds_load_b64 with conflict-free banking.
// ---------------------------------------------------------------------------
template <int COUT>
__global__ __launch_bounds__(256) void gcn_gemm_f32wmma(
    const float* __restrict__ X, const float* __restrict__ W,
    const float* __restrict__ bias, float* __restrict__ Y, int nrows) {
  constexpr int CIN  = 128;
  constexpr int NW   = COUT / 16;         // waves per block (8 or 4)
  constexpr int ROWS = 32;                // row tiles: 2 per wave
  __shared__ float As[ROWS * 132];
  __shared__ float Wt[COUT * 134];        // transposed: Wt[c][k]

  const int tid  = threadIdx.x;           // blockDim.x == NW*32
  const int wave = tid >> 5;
  const int lane = tid & 31;
  const int row0 = blockIdx.x * ROWS;
  constexpr int NT = NW * 32;

  // Stage 32x128 A panel (coalesced; zero-pad tail rows).
  for (int i = tid; i < ROWS * CIN; i += NT) {
    const int r = i >> 7;
    const int c = i & 127;
    float v = 0.0f;
    if (row0 + r < nrows) v = X[(size_t)(row0 + r) * CIN + c];
    As[r * 132 + c] = v;
  }
  // Stage W transposed (coalesced global reads; conflict-free LDS writes).
  for (int i = tid; i < CIN * COUT; i += NT) {
    const int k = i / COUT;
    const int c = i % COUT;               // consecutive tid -> consecutive c
    Wt[c * 134 + k] = W[i];
  }
  __syncthreads();

  const int m    = lane & 15;             // A row / B col within tile
  const int half = lane >> 4;             // selects k sub-pair {0,1} vs {2,3}
  const int col  = wave * 16 + m;         // global output column

  v8f acc0 = {}, acc1 = {};
#pragma unroll
  for (int k = 0; k < CIN; k += 4) {
    const int kk = k + half * 2;
    v2f b, a0, a1;
    b.x  = Wt[col * 134 + kk];
    b.y  = Wt[col * 134 + kk + 1];
    a0.x = As[m * 132 + kk];
    a0.y = As[m * 132 + kk + 1];
    a1.x = As[(m + 16) * 132 + kk];
    a1.y = As[(m + 16) * 132 + kk + 1];
    // EXEC is all-1s here (no divergence wraps these calls).
    acc0 = __builtin_amdgcn_wmma_f32_16x16x4_f32(
        false, a0, false, b, (short)0, acc0, false, false);
    acc1 = __builtin_amdgcn_wmma_f32_16x16x4_f32(
        false, a1, false, b, (short)0, acc1, false, false);
  }

  const float bv = bias ? bias[col] : 0.0f;
#pragma unroll
  for (int v = 0; v < 8; ++v) {
    const int r0 = row0 + v + 8 * half;   // C/D layout: vgpr v -> row v+8*half
    const int r1 = r0 + 16;
    if (r0 < nrows) Y[(size_t)r0 * COUT + col] = acc0[v] + bv;
    if (r1 < nrows) Y[(size_t)r1 * COUT + col] = acc1[v] + bv;
  }
}

// ---------------------------------------------------------------------------
// Degree / normalization kernels. deg includes self-loop (init 1.0).
// ---------------------------------------------------------------------------
__global__ void gcn_deg_init(float* __restrict__ deg, int n) {
  int i = blockIdx.x * blockDim.x + threadIdx.x;
  if (i < n) deg[i] = 1.0f;               // self-loop contribution
}

__global__ void gcn_deg_count(const int* __restrict__ ei, int E,
                              float* __restrict__ deg, int n) {
  int e = blockIdx.x * blockDim.x + threadIdx.x;
  if (e < E) {
    int d = ei[E + e];                    // edge_index[1] = dst
    if (d >= 0 && d < n) atomicAdd(&deg[d], 1.0f);
  }
}

__global__ void gcn_deg_to_dis(float* __restrict__ dis, int n) {
  int i = blockIdx.x * blockDim.x + threadIdx.x;
  if (i < n) dis[i] = rsqrtf(fmaxf(dis[i], 1.0f));  // deg>=1 always
}

// agg = hw * dis^2 (self-loop) + bias  -- fully initializes agg buffer.
__global__ void gcn_agg_init(const float* __restrict__ hw,
                             const float* __restrict__ dis,
                             const float* __restrict__ bias,
                             float* __restrict__ agg, int n) {
  size_t i = (size_t)blockIdx.x * blockDim.x + threadIdx.x;
  if (i < (size_t)n * 128) {
    int node = (int)(i >> 7);
    int c    = (int)(i & 127);
    float d  = dis[node];
    agg[i] = hw[i] * d * d + bias[c];
  }
}

// One wave per edge: lane carries 4 channels (float4), atomic adds into dst.
__global__ void gcn_edge_scatter(const float* __restrict__ hw,
                                 const float* __restrict__ dis,
                                 const int* __restrict__ ei,
                                 float* __restrict__ agg, int E, int n) {
  int e = blockIdx.x * (blockDim.x >> 5) + (threadIdx.x >> 5);
  if (e >= E) return;
  const int lane = threadIdx.x & 31;
  int s = ei[e];                          // edge_index[0] = src
  int d = ei[E + e];                      // edge_index[1] = dst
  if (s < 0 || s >= n || d < 0 || d >= n) return;
  const float w = dis[s] * dis[d];
  const float4 v = ((const float4*)(hw + (size_t)s * 128))[lane];
  float* a = agg + (size_t)d * 128 + lane * 4;
  atomicAdd(a + 0, v.x * w);
  atomicAdd(a + 1, v.y * w);
  atomicAdd(a + 2, v.z * w);
  atomicAdd(a + 3, v.w * w);
}

__global__ void gcn_relu(const float* __restrict__ in, float* __restrict__ out,
                         size_t n) {
  size_t i = (size_t)blockIdx.x * blockDim.x + threadIdx.x;
  if (i < n) out[i] = fmaxf(in[i], 0.0f);
}

// ---------------------------------------------------------------------------
extern "C" void kernel_launch(void* const* d_in, const int* in_sizes, int n_in,
                              void* d_out, int out_size, void* d_ws, size_t ws_size,
                              hipStream_t stream) {
  const float* x  = (const float*)d_in[0];
  const int*   ei = (const int*)  d_in[1];   // [2, E] int32
  const float* Wp = (const float*)d_in[2];   // [128,128]
  const float* bp = (const float*)d_in[3];   // [128]
  const float* Wc = (const float*)d_in[4];   // [3,128,128]
  const float* bc = (const float*)d_in[5];   // [3,128]
  const float* Wo = (const float*)d_in[6];   // [128,64]
  const float* bo = (const float*)d_in[7];   // [64]
  float* out = (float*)d_out;

  const int N = in_sizes[0] / 128;
  const int E = in_sizes[1] / 2;

  // Workspace layout: dis[N] | h[N*128] | hw[N*128] | agg[N*128]  (~77 MB)
  float* ws   = (float*)d_ws;
  size_t NA   = ((size_t)N + 127) & ~(size_t)127;
  float* dis  = ws;
  float* h    = ws + NA;
  float* hw   = h  + (size_t)N * 128;
  float* agg  = hw + (size_t)N * 128;

  const int tpb   = 256;
  const int gN    = (N + tpb - 1) / tpb;
  const int gE    = (E + tpb - 1) / tpb;
  const int gNC   = (int)(((size_t)N * 128 + tpb - 1) / tpb);
  const int gTile = (N + 31) / 32;          // 32 rows per block
  const int gEdgeWave = (E + 7) / 8;        // 8 waves (edges) per 256-thr block

  // Degree + symmetric normalization
  gcn_deg_init  <<<gN, tpb, 0, stream>>>(dis, N);
  gcn_deg_count <<<gE, tpb, 0, stream>>>(ei, E, dis, N);
  gcn_deg_to_dis<<<gN, tpb, 0, stream>>>(dis, N);

  // h = x @ Wp + bp
  gcn_gemm_f32wmma<128><<<gTile, 256, 0, stream>>>(x, Wp, bp, h, N);

  for (int l = 0; l < 3; ++l) {
    // hw = h @ Wc[l]
    gcn_gemm_f32wmma<128><<<gTile, 256, 0, stream>>>(
        h, Wc + (size_t)l * 128 * 128, nullptr, hw, N);
    // agg = hw*dis^2 + bc[l]; then scatter edges; then h = relu(agg)
    gcn_agg_init<<<gNC, tpb, 0, stream>>>(hw, dis, bc + l * 128, agg, N);
    gcn_edge_scatter<<<gEdgeWave, tpb, 0, stream>>>(hw, dis, ei, agg, E, N);
    gcn_relu<<<gNC, tpb, 0, stream>>>(agg, h, (size_t)N * 128);
  }

  // out = h @ Wo + bo   (COUT=64 -> 4 waves / block)
  gcn_gemm_f32wmma<64><<<gTile, 128, 0, stream>>>(h, Wo, bo, out, N);
}